// Framewise_16922171146748
// MI455X (gfx1250) — compile-verified
//
#include <hip/hip_runtime.h>

typedef __attribute__((ext_vector_type(2))) float v2f;
typedef __attribute__((ext_vector_type(8))) float v8f;

constexpr int B  = 32;
constexpr int C  = 256;
constexpr int T  = 8192;
constexpr int NW = 512;

// -----------------------------------------------------------------------------
// Kernel 1: scores[b,t] = sum_c features[b,c,t] * W[c] + bias
// via V_WMMA_F32_16X16X4_F32.  N dim = 16 consecutive t values, K dim = c
// (64 chained K=4 WMMAs), W replicated across the 16 M rows (all D rows equal;
// we read row 0, which lives in acc[0] on lanes 0..15).
//
// Fragment layout assumptions (ISA 7.12.2):
//   A 16x4 f32 : lanes 0-15 -> (K=0 in v0, K=1 in v1); lanes 16-31 -> (K=2,K=3)
//   B 4x16 f32 : v0 = row K=0 (lanes 0-15) / row K=2 (lanes 16-31)
//                v1 = row K=1 (lanes 0-15) / row K=3 (lanes 16-31)
// Each wave owns 4 tiles of 16 t (64 t), block = 8 waves = 512 t.
// -----------------------------------------------------------------------------
__global__ __launch_bounds__(256) void scores_wmma_kernel(
    const float* __restrict__ features, const float* __restrict__ W,
    const float* __restrict__ bias, float* __restrict__ scores)
{
  __shared__ float ldsW[C];
  const int tid = threadIdx.x;
  ldsW[tid] = W[tid];               // C == blockDim.x == 256
  __syncthreads();

  const int blocksPerB = T / 512;
  const int b    = blockIdx.x / blocksPerB;
  const int tb   = (blockIdx.x % blocksPerB) * 512;
  const int wave = tid >> 5;
  const int lane = tid & 31;
  const int half = lane >> 4;       // 0: K=0,1   1: K=2,3
  const int tl   = lane & 15;       // N index within a tile
  const int t0   = tb + wave * 64;

  const float* __restrict__ fb = features + (size_t)b * C * T;

  v8f acc0 = {}, acc1 = {}, acc2 = {}, acc3 = {};

  #pragma unroll 4
  for (int c0 = 0; c0 < C; c0 += 4) {
    const int row = c0 + 2 * half;                 // this half's first K row
    v2f a = *(const v2f*)&ldsW[row];               // (W[row], W[row+1]) - 8B aligned
    const float* __restrict__ fr = fb + (size_t)row * T + t0 + tl;

    // prefetch 8 c-steps ahead (streams the next feature rows into cache)
    __builtin_prefetch(fr + (size_t)8 * T, 0, 0);

    v2f b0 = { fr[ 0], fr[(size_t)T +  0] };
    v2f b1 = { fr[16], fr[(size_t)T + 16] };
    v2f b2 = { fr[32], fr[(size_t)T + 32] };
    v2f b3 = { fr[48], fr[(size_t)T + 48] };

    acc0 = __builtin_amdgcn_wmma_f32_16x16x4_f32(false, a, false, b0, (short)0, acc0, false, false);
    acc1 = __builtin_amdgcn_wmma_f32_16x16x4_f32(false, a, false, b1, (short)0, acc1, false, false);
    acc2 = __builtin_amdgcn_wmma_f32_16x16x4_f32(false, a, false, b2, (short)0, acc2, false, false);
    acc3 = __builtin_amdgcn_wmma_f32_16x16x4_f32(false, a, false, b3, (short)0, acc3, false, false);
  }

  const float bv = bias[0];
  if (lane < 16) {                  // D row 0 = scores for this tile, N = lane
    float* __restrict__ so = scores + (size_t)b * T + t0 + tl;
    so[ 0] = acc0[0] + bv;
    so[16] = acc1[0] + bv;
    so[32] = acc2[0] + bv;
    so[48] = acc3[0] + bv;
  }
}

// -----------------------------------------------------------------------------
// Kernel 2: per-batch exclusive cumsum -> cs[b][0..T], cs[0]=0, cs[p]=sum s[0..p-1]
// One block per batch row; 256 threads x 32 elements each; block scan in LDS.
// -----------------------------------------------------------------------------
__global__ __launch_bounds__(256) void scan_kernel(
    const float* __restrict__ scores, float* __restrict__ cs)
{
  const int b   = blockIdx.x;
  const int tid = threadIdx.x;
  constexpr int PER = T / 256;      // 32
  const float* __restrict__ s = scores + (size_t)b * T;
  float*       __restrict__ c = cs     + (size_t)b * (T + 1);

  float v[PER];
  float sum = 0.f;
  const int base = tid * PER;
  #pragma unroll
  for (int i = 0; i < PER; ++i) { v[i] = s[base + i]; sum += v[i]; }

  __shared__ float l[256];
  l[tid] = sum;
  __syncthreads();
  #pragma unroll
  for (int off = 1; off < 256; off <<= 1) {
    float t = (tid >= off) ? l[tid - off] : 0.f;
    __syncthreads();
    l[tid] += t;
    __syncthreads();
  }
  float run = l[tid] - sum;         // exclusive prefix of this thread's chunk

  if (tid == 0) c[0] = 0.f;
  #pragma unroll
  for (int i = 0; i < PER; ++i) { run += v[i]; c[base + i + 1] = run; }
}

// -----------------------------------------------------------------------------
// Kernel 3: word_scores[b,w] = (cs[e] - cs[s]) / (e - s)
// word_bounds layout (B,2,NW): starts at [b*2*NW + w], ends at [b*2*NW + NW + w]
// -----------------------------------------------------------------------------
__global__ void word_kernel(const float* __restrict__ cs,
                            const int* __restrict__ wb,
                            float* __restrict__ out)
{
  const int idx = blockIdx.x * blockDim.x + threadIdx.x;
  if (idx >= B * NW) return;
  const int b = idx / NW;
  const int w = idx - b * NW;
  const int s = wb[(size_t)b * 2 * NW + w];
  const int e = wb[(size_t)b * 2 * NW + NW + w];
  const float* __restrict__ c = cs + (size_t)b * (T + 1);
  out[idx] = (c[e] - c[s]) / (float)(e - s);
}

// -----------------------------------------------------------------------------
// Inputs (setup_inputs order): features f32[B,C,T], W f32[C], b f32[1],
// word_bounds i32[B,2,NW], word_lengths i32[B] (unused by reference).
// Output: f32[B,1,NW].  Workspace: scores (B*T f32) + cs (B*(T+1) f32) ~2.1 MB.
// -----------------------------------------------------------------------------
extern "C" void kernel_launch(void* const* d_in, const int* in_sizes, int n_in,
                              void* d_out, int out_size, void* d_ws, size_t ws_size,
                              hipStream_t stream)
{
  (void)in_sizes; (void)n_in; (void)out_size; (void)ws_size;
  const float* features = (const float*)d_in[0];
  const float* W        = (const float*)d_in[1];
  const float* bias     = (const float*)d_in[2];
  const int*   wb       = (const int*)d_in[3];

  float* scores = (float*)d_ws;
  float* cs     = scores + (size_t)B * T;

  scores_wmma_kernel<<<B * (T / 512), 256, 0, stream>>>(features, W, bias, scores);
  scan_kernel<<<B, 256, 0, stream>>>(scores, cs);
  word_kernel<<<(B * NW + 255) / 256, 256, 0, stream>>>(cs, wb, (float*)d_out);
}